// VmfVectorQuantizer_27504970563602
// MI455X (gfx1250) — compile-verified
//
#include <hip/hip_runtime.h>
#include <math.h>

typedef float v2f __attribute__((ext_vector_type(2)));
typedef float v8f __attribute__((ext_vector_type(8)));

#define N_ROWS 32768
#define DZ 256
#define KD 512
#define ROWS_PER_WG 32
#define ZT_STRIDE 258   // padded: (258 % 64) == 2 -> conflict-free strided lane reads
#define LG_STRIDE 514   // padded: (514 % 64) == 2

// ---------------------------------------------------------------------------
// Row-group reductions: 8 threads per row (r = t>>3, j = t&7), scratch red[256]
// ---------------------------------------------------------------------------
__device__ __forceinline__ float wg_row_reduce_max(float* red, int r, int j, float v) {
    red[r * 8 + j] = v;
    __syncthreads();
    float m = red[r * 8 + 0];
#pragma unroll
    for (int k = 1; k < 8; ++k) m = fmaxf(m, red[r * 8 + k]);
    __syncthreads();
    return m;
}

__device__ __forceinline__ float wg_row_reduce_sum(float* red, int r, int j, float v) {
    red[r * 8 + j] = v;
    __syncthreads();
    float s = 0.f;
#pragma unroll
    for (int k = 0; k < 8; ++k) s += red[r * 8 + k];
    __syncthreads();
    return s;
}

// ---------------------------------------------------------------------------
// Kernel 1: L2-normalize codebook rows into ws; block 0 zeroes accumulators.
// ---------------------------------------------------------------------------
__global__ __launch_bounds__(256) void cb_norm_kernel(const float* __restrict__ cb,
                                                      float* __restrict__ cbn,
                                                      float* __restrict__ avg_sum,
                                                      float* __restrict__ loss_acc) {
    __shared__ float red[256];
    const int row = blockIdx.x;
    const int t = threadIdx.x;
    const float v = cb[row * DZ + t];
    red[t] = v * v;
    __syncthreads();
    for (int s = 128; s > 0; s >>= 1) {
        if (t < s) red[t] += red[t + s];
        __syncthreads();
    }
    const float norm = sqrtf(red[0]);
    cbn[row * DZ + t] = v / norm;

    if (row == 0) {
        avg_sum[t] = 0.f;
        avg_sum[t + 256] = 0.f;
        if (t == 0) loss_acc[0] = 0.f;
    }
}

// ---------------------------------------------------------------------------
// Kernel 2: fused VQ kernel. 1 workgroup = 32 token rows, 256 threads = 8 waves.
//   stage 1: gather z tile (transposed load) into LDS
//   stage 2: logits = kq * z @ cbn^T via v_wmma_f32_16x16x4_f32 (K=256)
//   stage 3: softmax / log-softmax / gumbel-softmax chain in LDS
//   stage 4: z_q = encodings @ cbn via v_wmma_f32_16x16x4_f32 (K=512),
//            write z_to_decoder, accumulate loss terms
// ---------------------------------------------------------------------------
__global__ __launch_bounds__(256) void vq_main_kernel(const float* __restrict__ z,
                                                      const float* __restrict__ kappa,
                                                      const float* __restrict__ gu,
                                                      const float* __restrict__ cbn,
                                                      float* __restrict__ out,
                                                      float* __restrict__ avg_sum,
                                                      float* __restrict__ loss_acc) {
    __shared__ float zt[ROWS_PER_WG * ZT_STRIDE];   // z tile   [row][d], padded
    __shared__ float lg[ROWS_PER_WG * LG_STRIDE];   // logits -> encodings [row][code]
    __shared__ float avgp[KD];                      // per-WG sum of probabilities
    __shared__ float red[256];                      // row-reduction scratch

    const int t  = threadIdx.x;
    const int w  = t >> 5;    // wave id 0..7
    const int l  = t & 31;    // lane id (wave32)
    const int lh = l >> 4;    // lane half (K-pair select for A/B fragments)
    const int ll = l & 15;    // lane-in-half (M or N index)

    const int n0  = blockIdx.x * ROWS_PER_WG;
    const int b   = n0 >> 10;       // batch index (1024 tokens per batch elem)
    const int wh0 = n0 & 1023;      // W*H offset (32-aligned)

    const float kq = fminf(fmaxf(kappa[0], 1e-5f), 1e5f);

    avgp[t] = 0.f;
    avgp[t + 256] = 0.f;

    // ---- stage 1: load z tile. z_flat[n,d] = z[(b*256+d)*1024 + (n&1023)] ----
    {
        const int r = t & 31;
        const int dbase = t >> 5;
#pragma unroll
        for (int i = 0; i < 32; ++i) {
            const int d = dbase + (i << 3);
            zt[r * ZT_STRIDE + d] = z[((b * DZ + d) << 10) + wh0 + r];
        }
    }
    __syncthreads();

    float thr_loss = 0.f;

    // ---- stage 2: logits[32][512] = kq * (z @ cbn^T), K = 256 ----
    // 64 output tiles (2 M-tiles x 32 code-tiles); wave w takes tiles w, w+8, ...
    for (int it = 0; it < 8; ++it) {
        const int tile = w + (it << 3);
        const int mt = tile >> 5;
        const int ct = tile & 31;
        const int m = mt * 16 + ll;
        const int code = ct * 16 + ll;
        const float* arow = &zt[m * ZT_STRIDE + 2 * lh];  // A: lane M=ll, K pair 2h,2h+1
        const float* brow = cbn + code * DZ + 2 * lh;     // B: lane N=ll, K pair 2h,2h+1
        v8f acc = {};
        for (int k0 = 0; k0 < DZ; k0 += 4) {
            v2f a  = *(const v2f*)(arow + k0);
            v2f bb = *(const v2f*)(brow + k0);
            acc = __builtin_amdgcn_wmma_f32_16x16x4_f32(false, a, false, bb,
                                                        (short)0, acc, false, false);
        }
#pragma unroll
        for (int g = 0; g < 8; ++g) {
            const int row = mt * 16 + g + 8 * lh;          // C/D: VGPR g, lane-half h
            lg[row * LG_STRIDE + ct * 16 + ll] = kq * acc[g];
        }
    }
    __syncthreads();

    // ---- stage 3: softmax chain; 8 threads per row, 64 cols each ----
    {
        const int r = t >> 3;
        const int j = t & 7;
        float* lrow = &lg[r * LG_STRIDE];
        const size_t nrow = (size_t)(n0 + r);

        // pass 1: row max
        float m = -1e30f;
        for (int i = 0; i < 64; ++i) m = fmaxf(m, lrow[j + (i << 3)]);
        m = wg_row_reduce_max(red, r, j, m);

        // pass 2: clipped shifted logit, sum of exp
        float s1 = 0.f;
        for (int i = 0; i < 64; ++i) {
            const int c = j + (i << 3);
            const float lc = fminf(fmaxf(lrow[c] - m, -50.f), 50.f);
            lrow[c] = lc;
            s1 += expf(lc);
        }
        s1 = wg_row_reduce_sum(red, r, j, s1);
        const float logS = logf(s1);
        const float inv1 = 1.f / s1;

        // pass 3: p, p*logp, avg-prob accumulate, gumbel perturbation q
        float m2 = -1e30f;
        for (int i = 0; i < 64; ++i) {
            const int c = j + (i << 3);
            const float lc = lrow[c];
            const float p = expf(lc) * inv1;
            thr_loss += p * (lc - logS);                  // kld_discrete numerator
            atomicAdd(&avgp[c], p);
            const float u = gu[nrow * KD + c];
            const float gmb = -logf(-logf(u + 1e-10f) + 1e-10f);
            const float q = (lc + gmb) * 2.0f;            // 1/TEMPERATURE
            lrow[c] = q;
            m2 = fmaxf(m2, q);
        }
        m2 = wg_row_reduce_max(red, r, j, m2);

        // pass 4: gumbel-softmax denominator
        float s2 = 0.f;
        for (int i = 0; i < 64; ++i) s2 += expf(lrow[j + (i << 3)] - m2);
        s2 = wg_row_reduce_sum(red, r, j, s2);
        const float inv2 = 1.f / s2;

        // pass 5: encodings into lg
        for (int i = 0; i < 64; ++i) {
            const int c = j + (i << 3);
            lrow[c] = expf(lrow[c] - m2) * inv2;
        }
    }
    __syncthreads();

    // flush per-WG probability sums
    atomicAdd(&avg_sum[t], avgp[t]);
    atomicAdd(&avg_sum[t + 256], avgp[t + 256]);

    // ---- stage 4: z_q[32][256] = encodings @ cbn, K = 512 ----
    // 32 output tiles (2 M-tiles x 16 D-tiles); wave w takes tiles w, w+8, ...
    for (int it = 0; it < 4; ++it) {
        const int tile = w + (it << 3);
        const int mt = tile >> 4;
        const int dt = tile & 15;
        const int m = mt * 16 + ll;
        const int dcol = dt * 16 + ll;
        const float* arow = &lg[m * LG_STRIDE + 2 * lh];
        const float* bcol = cbn + (2 * lh) * DZ + dcol;   // B[k][d]: cbn[k*256 + d]
        v8f acc = {};
        for (int k0 = 0; k0 < KD; k0 += 4) {
            v2f a = *(const v2f*)(arow + k0);
            v2f bb;
            bb.x = bcol[k0 * DZ];
            bb.y = bcol[k0 * DZ + DZ];
            acc = __builtin_amdgcn_wmma_f32_16x16x4_f32(false, a, false, bb,
                                                        (short)0, acc, false, false);
        }
#pragma unroll
        for (int g = 0; g < 8; ++g) {
            const int row = mt * 16 + g + 8 * lh;
            const float zq = acc[g];
            const float zv = zt[row * ZT_STRIDE + dcol];
            thr_loss += kq * zv * (zv - zq);               // kld_continuous numerator
            out[((b * DZ + dcol) << 10) + wh0 + row] = zq; // z_to_decoder [b][d][w][h]
        }
    }

    atomicAdd(loss_acc, thr_loss * (1.0f / 32.0f));        // both terms are /bs
}

// ---------------------------------------------------------------------------
// Kernel 3: finalize loss + perplexity scalars.
// ---------------------------------------------------------------------------
__global__ __launch_bounds__(512) void finalize_kernel(const float* __restrict__ avg_sum,
                                                       const float* __restrict__ loss_acc,
                                                       float* __restrict__ out) {
    __shared__ float red[512];
    const int t = threadIdx.x;
    float a = avg_sum[t] * (1.0f / (float)N_ROWS);
    a = fmaxf(a, 1e-10f);
    red[t] = a * logf(a + 1e-10f);
    __syncthreads();
    for (int s = 256; s > 0; s >>= 1) {
        if (t < s) red[t] += red[t + s];
        __syncthreads();
    }
    if (t == 0) {
        out[N_ROWS * DZ]     = loss_acc[0];
        out[N_ROWS * DZ + 1] = expf(-red[0]);
    }
}

// ---------------------------------------------------------------------------
extern "C" void kernel_launch(void* const* d_in, const int* in_sizes, int n_in,
                              void* d_out, int out_size, void* d_ws, size_t ws_size,
                              hipStream_t stream) {
    (void)in_sizes; (void)n_in; (void)out_size; (void)ws_size;

    const float* z     = (const float*)d_in[0];   // [32,256,32,32]
    const float* kappa = (const float*)d_in[1];   // [1]
    const float* cb    = (const float*)d_in[2];   // [512,256]
    const float* gu    = (const float*)d_in[3];   // [32768,512]
    float* out = (float*)d_out;                   // 32*256*32*32 + 2 floats

    float* cbn      = (float*)d_ws;               // 512*256 floats
    float* avg_sum  = cbn + KD * DZ;              // 512 floats
    float* loss_acc = avg_sum + KD;               // 1 float

    cb_norm_kernel<<<KD, 256, 0, stream>>>(cb, cbn, avg_sum, loss_acc);
    vq_main_kernel<<<N_ROWS / ROWS_PER_WG, 256, 0, stream>>>(z, kappa, gu, cbn, out,
                                                             avg_sum, loss_acc);
    finalize_kernel<<<1, 512, 0, stream>>>(avg_sum, loss_acc, out);
}